// VQBottleneck_56410100465700
// MI455X (gfx1250) — compile-verified
//
#include <hip/hip_runtime.h>
#include <hip/hip_bf16.h>

typedef __attribute__((ext_vector_type(2))) float v2f;
typedef __attribute__((ext_vector_type(8))) float v8f;

#define B_      8
#define T_      2048
#define IN_DIM  1024
#define LATENT  256
#define NTOK    4096
#define NROWS   (B_ * T_)          // 16384
#define NEG_INF (-3.402823466e+38f)

// ---------------------------------------------------------------------------
// Kernel 0: cnorm_half[k] = 0.5 * |codebook[k]|^2   (one wave32 per row)
// ---------------------------------------------------------------------------
__global__ void __launch_bounds__(256) vq_cnorm(const float* __restrict__ cb,
                                                float* __restrict__ cnh) {
    const int wave = (blockIdx.x * blockDim.x + threadIdx.x) >> 5;
    const int lane = threadIdx.x & 31;
    const float* r = cb + (size_t)wave * LATENT;
    float s = 0.f;
#pragma unroll
    for (int j = 0; j < LATENT / 32; ++j) {
        float v = r[lane + 32 * j];
        s += v * v;
    }
#pragma unroll
    for (int off = 16; off >= 1; off >>= 1) s += __shfl_xor(s, off, 32);
    if (lane == 0) cnh[wave] = 0.5f * s;
}

// ---------------------------------------------------------------------------
// Kernel A: h = x @ W + b   (M=16384, K=1024, N=256), fp32 WMMA 16x16x4.
// One wave per 16x16 output tile; grid = (M/16, N/64), 4 waves/block.
// h is staged in the `codes` region of d_out (same 16384x256 f32 shape);
// the final gather kernel fully overwrites it.
// ---------------------------------------------------------------------------
__global__ void __launch_bounds__(128) vq_proj(const float* __restrict__ x,
                                               const float* __restrict__ W,
                                               const float* __restrict__ bias,
                                               float* __restrict__ h) {
    const int wave  = threadIdx.x >> 5;
    const int lane  = threadIdx.x & 31;
    const int n16   = lane & 15;
    const int half  = lane >> 4;
    const int kk    = half * 2;
    const int mtile = blockIdx.x;
    const int ntile = blockIdx.y * 4 + wave;
    const int ncol  = ntile * 16 + n16;
    const int arow  = mtile * 16 + n16;

    const float* xrow = x + (size_t)arow * IN_DIM + kk;

    v8f d = {0.f, 0.f, 0.f, 0.f, 0.f, 0.f, 0.f, 0.f};
    for (int k = 0; k < IN_DIM; k += 4) {
        v2f a;
        {
            const float2 t = *(const float2*)(xrow + k);
            a.x = t.x; a.y = t.y;
        }
        v2f b;
        b.x = W[(size_t)(k + kk) * LATENT + ncol];
        b.y = W[(size_t)(k + kk + 1) * LATENT + ncol];
        d = __builtin_amdgcn_wmma_f32_16x16x4_f32(false, a, false, b,
                                                  (short)0, d, false, false);
    }

    const float bv = bias[ncol];
    const int rowBase = mtile * 16 + half * 8;
#pragma unroll
    for (int r = 0; r < 8; ++r)
        h[(size_t)(rowBase + r) * LATENT + ncol] = d[r] + bv;
}

// ---------------------------------------------------------------------------
// Kernel B: scores = h @ cb^T - 0.5|cb|^2, fused per-row argmax (== argmin d2).
// One wave per 16-row strip; A-strip (16x256) kept in 128 VGPRs, reused over
// all 256 codebook tiles. -0.5|cb|^2 folded into WMMA C-init.
// ---------------------------------------------------------------------------
__global__ void __launch_bounds__(128) vq_argmin(const float* __restrict__ h,
                                                 const float* __restrict__ cb,
                                                 const float* __restrict__ cnh,
                                                 int* __restrict__ idx,
                                                 float* __restrict__ idxf) {
    const int wave  = threadIdx.x >> 5;
    const int lane  = threadIdx.x & 31;
    const int n16   = lane & 15;
    const int half  = lane >> 4;
    const int kk    = half * 2;
    const int mtile = blockIdx.x * 4 + wave;
    const int arow  = mtile * 16 + n16;

    // Preload the 16x256 h-strip into registers (WMMA A layout, 64 k-steps).
    v2f areg[64];
    const float* hrow = h + (size_t)arow * LATENT + kk;
#pragma unroll
    for (int ks = 0; ks < 64; ++ks) {
        const float2 t = *(const float2*)(hrow + ks * 4);
        areg[ks].x = t.x; areg[ks].y = t.y;
    }

    float best[8];
    int   bidx[8];
#pragma unroll
    for (int r = 0; r < 8; ++r) { best[r] = NEG_INF; bidx[r] = 0; }

    for (int nt = 0; nt < NTOK / 16; ++nt) {
        const int col = nt * 16 + n16;
        const float cin = -cnh[col];                 // fold -0.5|cb|^2 into C
        v8f d = {cin, cin, cin, cin, cin, cin, cin, cin};
        const float* crow = cb + (size_t)col * LATENT + kk;  // cb^T tile
#pragma unroll
        for (int ks = 0; ks < 64; ++ks) {
            v2f b;
            const float2 t = *(const float2*)(crow + ks * 4);
            b.x = t.x; b.y = t.y;
            d = __builtin_amdgcn_wmma_f32_16x16x4_f32(false, areg[ks], false, b,
                                                      (short)0, d, false, false);
        }
#pragma unroll
        for (int r = 0; r < 8; ++r)
            if (d[r] > best[r]) { best[r] = d[r]; bidx[r] = col; }  // strict >: first-hit wins
    }

    // Per-row (val,idx) argmax reduction across the 16 lanes of each half-wave.
#pragma unroll
    for (int r = 0; r < 8; ++r) {
        float v = best[r];
        int   bi = bidx[r];
#pragma unroll
        for (int off = 8; off >= 1; off >>= 1) {
            float ov = __shfl_xor(v, off, 16);
            int   oi = __shfl_xor(bi, off, 16);
            if (ov > v || (ov == v && oi < bi)) { v = ov; bi = oi; }
        }
        if (n16 == 0) {
            const int row = mtile * 16 + half * 8 + r;
            idx[row]  = bi;
            idxf[row] = (float)bi;
        }
    }
}

// ---------------------------------------------------------------------------
// Kernel C: codes[row] = codebook[idx[row]]  (float4 = B128 traffic).
// Fully overwrites the codes region (which served as h staging).
// ---------------------------------------------------------------------------
__global__ void __launch_bounds__(256) vq_gather(const int* __restrict__ idx,
                                                 const float* __restrict__ cb,
                                                 float* __restrict__ codes) {
    const int t = blockIdx.x * blockDim.x + threadIdx.x;  // NROWS * 64 threads
    const int row = t >> 6;
    const int j   = t & 63;
    const float4* cb4 = (const float4*)cb;
    float4* out4 = (float4*)codes;
    out4[(size_t)row * 64 + j] = cb4[(size_t)idx[row] * 64 + j];
}

// ---------------------------------------------------------------------------
extern "C" void kernel_launch(void* const* d_in, const int* in_sizes, int n_in,
                              void* d_out, int out_size, void* d_ws, size_t ws_size,
                              hipStream_t stream) {
    const float* x    = (const float*)d_in[0];  // [8,2048,1024]
    const float* W    = (const float*)d_in[1];  // [1024,256]
    const float* bias = (const float*)d_in[2];  // [256]
    const float* cb   = (const float*)d_in[3];  // [4096,256]

    float* out   = (float*)d_out;
    float* codes = out;                               // NROWS*LATENT floats
    float* idxf  = out + (size_t)NROWS * LATENT;      // NROWS values (as float)
    float* h     = codes;                             // stage h in codes region

    // Workspace: only 80 KB needed (cnh + int indices).
    char*  ws  = (char*)d_ws;
    float* cnh = (float*)ws;                                   // 16 KB
    int*   idx = (int*)(ws + (size_t)NTOK * sizeof(float));    // 64 KB

    // 0) half codebook norms: 4096 rows, 8 waves/block
    vq_cnorm<<<NTOK / 8, 256, 0, stream>>>(cb, cnh);

    // A) projection GEMM: grid (M/16 tiles, N/64), 4 waves per block
    dim3 gA(NROWS / 16, LATENT / 64);
    vq_proj<<<gA, 128, 0, stream>>>(x, W, bias, h);

    // B) distance GEMM + fused argmin: 4 M-tiles per block
    vq_argmin<<<(NROWS / 16) / 4, 128, 0, stream>>>(h, cb, cnh, idx, idxf);

    // C) gather codes (overwrites h staging with final codes)
    vq_gather<<<(NROWS * 64) / 256, 256, 0, stream>>>(idx, cb, codes);
}